// FwdOnlineLRU_85804856639685
// MI455X (gfx1250) — compile-verified
//
#include <hip/hip_runtime.h>
#include <hip/hip_bf16.h>
#include <math.h>

typedef __attribute__((ext_vector_type(2))) float v2f;
typedef __attribute__((ext_vector_type(4))) float v4f;
typedef __attribute__((ext_vector_type(8))) float v8f;

#define BATCH 256
#define HID   256
#define DIN   256

// ---------------------------------------------------------------------------
// Kernel 0: precompute per-h constants once (256 entries):
//   lam_re[h] = exp(-exp(nu)) * cos(exp(theta))
//   lam_im[h] = exp(-exp(nu)) * sin(exp(theta))
//   eg[h]     = exp(gamma_log)
// Removes ~170M redundant transcendental VALU ops from the streaming kernel.
// ---------------------------------------------------------------------------
__global__ __launch_bounds__(256) void lru_lambda_precompute_kernel(
    const float* __restrict__ nu_log,
    const float* __restrict__ theta_log,
    const float* __restrict__ gamma_log,
    float* __restrict__ lam_tab)          // [3*HID]: lam_re | lam_im | eg
{
    const int h = threadIdx.x;            // HID == 256 == blockDim.x
    const float mod = expf(-expf(nu_log[h]));
    const float th  = expf(theta_log[h]);
    lam_tab[0 * HID + h] = mod * cosf(th);
    lam_tab[1 * HID + h] = mod * sinf(th);
    lam_tab[2 * HID + h] = expf(gamma_log[h]);
}

// ---------------------------------------------------------------------------
// Kernel 1: complex GEMM P = u @ Bc^T via V_WMMA_F32_16X16X4_F32 (fp32 WMMA),
// fused with the Lambda-recurrence epilogue producing out1 = (6, B, H).
// One wave32 per 16x16 output tile; 8 waves (256 threads) per block.
// ---------------------------------------------------------------------------
__global__ __launch_bounds__(256) void lru_state_wmma_kernel(
    const float* __restrict__ h_re,  const float* __restrict__ h_im,
    const float* __restrict__ gl_re, const float* __restrict__ gl_im,
    const float* __restrict__ gg_re, const float* __restrict__ gg_im,
    const float* __restrict__ u,          // (B, D)
    const float* __restrict__ lam_tab,    // [3*HID]
    const float* __restrict__ B_re,       // (H, D)
    const float* __restrict__ B_im,       // (H, D)
    float* __restrict__ out1)             // (6, B, H)
{
    const int lane = threadIdx.x & 31;
    const int wave = threadIdx.x >> 5;
    const int tile = blockIdx.x * 8 + wave;           // 0..255
    const int tiles_h = HID / 16;                     // 16
    const int b0 = (tile / tiles_h) * 16;
    const int h0 = (tile % tiles_h) * 16;

    const int half = lane >> 4;                       // 0|1 -> K pair {0,1} / {2,3}
    const int l15  = lane & 15;

    // ISA 7.12.2: 32-bit A 16x4 -> lane holds row m=l15, VGPR0=K(2*half), VGPR1=K(2*half+1)
    // B 4x16 mirrors: lane holds col n=l15, same K pairing.  Both contiguous in d.
    const float* a_ptr  = u    + (size_t)(b0 + l15) * DIN + half * 2;
    const float* br_ptr = B_re + (size_t)(h0 + l15) * DIN + half * 2;
    const float* bi_ptr = B_im + (size_t)(h0 + l15) * DIN + half * 2;

    v8f c_re = {0.f, 0.f, 0.f, 0.f, 0.f, 0.f, 0.f, 0.f};
    v8f c_im = {0.f, 0.f, 0.f, 0.f, 0.f, 0.f, 0.f, 0.f};

#pragma unroll 8
    for (int k = 0; k < DIN; k += 4) {
        v2f a  = *(const v2f*)(a_ptr  + k);
        v2f br = *(const v2f*)(br_ptr + k);
        v2f bi = *(const v2f*)(bi_ptr + k);
        // 8 args: (neg_a, A, neg_b, B, c_mod, C, reuse_a, reuse_b)
        c_re = __builtin_amdgcn_wmma_f32_16x16x4_f32(
            false, a, false, br, (short)0, c_re, false, false);
        c_im = __builtin_amdgcn_wmma_f32_16x16x4_f32(
            false, a, false, bi, (short)0, c_im, false, false);
    }

    // Per-lane h index is fixed: h = h0 + (lane & 15)
    const int h   = h0 + l15;
    const float Lre = lam_tab[0 * HID + h];
    const float Lim = lam_tab[1 * HID + h];
    const float eg  = lam_tab[2 * HID + h];

    const int BH = BATCH * HID;
#pragma unroll
    for (int r = 0; r < 8; ++r) {
        // C/D layout: VGPR r -> M=r (lanes 0-15), M=8+r (lanes 16-31); N = l15
        const int b   = b0 + r + half * 8;
        const int idx = b * HID + h;
        const float pre = c_re[r];
        const float pim = c_im[r];
        const float hr  = h_re[idx],  hi  = h_im[idx];
        const float glr = gl_re[idx], gli = gl_im[idx];
        const float ggr = gg_re[idx], ggi = gg_im[idx];
        out1[0 * BH + idx] = Lre * hr  - Lim * hi  + eg * pre;  // h_t.real
        out1[1 * BH + idx] = Lre * hi  + Lim * hr  + eg * pim;  // h_t.imag
        out1[2 * BH + idx] = Lre * glr - Lim * gli + hr;        // gl_t.real
        out1[3 * BH + idx] = Lre * gli + Lim * glr + hi;        // gl_t.imag
        out1[4 * BH + idx] = Lre * ggr - Lim * ggi + pre;       // gg_t.real
        out1[5 * BH + idx] = Lre * ggi + Lim * ggr + pim;       // gg_t.imag
    }
}

// ---------------------------------------------------------------------------
// Kernel 2: gB_t = Lam * gB + gamma_log[h] * u[b,d]   (B,H,D) complex.
// Pure 268 MB HBM stream -> float4 per thread, nontemporal loads/stores.
// Per-h constants come from the precomputed table (L1-resident broadcast).
// ---------------------------------------------------------------------------
__global__ __launch_bounds__(256) void lru_gB_stream_kernel(
    const float* __restrict__ gB_re, const float* __restrict__ gB_im,
    const float* __restrict__ u,          // (B, D)
    const float* __restrict__ lam_tab,    // [3*HID]
    const float* __restrict__ gamma_log,  // (1, H)
    float* __restrict__ o_re,             // (B, H, D)
    float* __restrict__ o_im)             // (B, H, D)
{
    const int idx4 = blockIdx.x * blockDim.x + threadIdx.x; // 0 .. B*H*D/4-1
    const int d4 = idx4 & (DIN / 4 - 1);
    const int h  = (idx4 >> 6) & (HID - 1);
    const int b  = idx4 >> 14;

    const float Lre = lam_tab[0 * HID + h];
    const float Lim = lam_tab[1 * HID + h];
    const float gam = gamma_log[h];

    const int base = idx4 * 4;            // flat (b,h,d) element offset
    const v4f gre = __builtin_nontemporal_load((const v4f*)(gB_re + base));
    const v4f gim = __builtin_nontemporal_load((const v4f*)(gB_im + base));
    const v4f uu  = *(const v4f*)(u + b * DIN + d4 * 4);

    const v4f ore = Lre * gre - Lim * gim + gam * uu;
    const v4f oim = Lre * gim + Lim * gre;

    __builtin_nontemporal_store(ore, (v4f*)(o_re + base));
    __builtin_nontemporal_store(oim, (v4f*)(o_im + base));
}

extern "C" void kernel_launch(void* const* d_in, const int* in_sizes, int n_in,
                              void* d_out, int out_size, void* d_ws, size_t ws_size,
                              hipStream_t stream) {
    // setup_inputs order:
    const float* h_re   = (const float*)d_in[0];
    const float* h_im   = (const float*)d_in[1];
    const float* gl_re  = (const float*)d_in[2];
    const float* gl_im  = (const float*)d_in[3];
    const float* gg_re  = (const float*)d_in[4];
    const float* gg_im  = (const float*)d_in[5];
    const float* gB_re  = (const float*)d_in[6];
    const float* gB_im  = (const float*)d_in[7];
    const float* inputs = (const float*)d_in[8];
    const float* nu_log    = (const float*)d_in[9];
    const float* theta_log = (const float*)d_in[10];
    const float* gamma_log = (const float*)d_in[11];
    const float* B_re   = (const float*)d_in[12];
    const float* B_im   = (const float*)d_in[13];

    float* out1 = (float*)d_out;                                 // (6,B,H)
    float* o_re = out1 + (size_t)6 * BATCH * HID;                // (B,H,D)
    float* o_im = o_re + (size_t)BATCH * HID * DIN;              // (B,H,D)

    float* lam_tab = (float*)d_ws;                               // 3*HID floats

    // Per-h Lambda table (3 KB) — written every call, fully deterministic.
    lru_lambda_precompute_kernel<<<1, 256, 0, stream>>>(
        nu_log, theta_log, gamma_log, lam_tab);

    // 256 tiles of 16x16, 8 waves/block -> 32 blocks
    lru_state_wmma_kernel<<<32, 256, 0, stream>>>(
        h_re, h_im, gl_re, gl_im, gg_re, gg_im, inputs,
        lam_tab, B_re, B_im, out1);

    // B*H*D/4 = 4,194,304 float4 threads -> 16384 blocks of 256
    lru_gB_stream_kernel<<<(BATCH * HID * DIN / 4) / 256, 256, 0, stream>>>(
        gB_re, gB_im, inputs, lam_tab, gamma_log, o_re, o_im);
}